// SGC_8409545965932
// MI455X (gfx1250) — compile-verified
//
#include <hip/hip_runtime.h>
#include <hip/hip_bf16.h>
#include <math.h>

#define N_NODES   100000
#define N_EDGES   600000
#define D_FEAT    128
#define N_CLASSES 40

typedef __attribute__((ext_vector_type(2))) float v2f;
typedef __attribute__((ext_vector_type(8))) float v8f;

// ---------------- degree / normalization ----------------

__global__ void sgc_deg_init(float* deg) {
    int i = blockIdx.x * blockDim.x + threadIdx.x;
    if (i < N_NODES) deg[i] = 1.0f;  // self loop contributes 1
}

__global__ void sgc_deg_accum(const int* dst, float* deg) {
    int e = blockIdx.x * blockDim.x + threadIdx.x;
    if (e < N_EDGES) atomicAdd(&deg[dst[e]], 1.0f);
}

__global__ void sgc_dinv(float* deg) {
    int i = blockIdx.x * blockDim.x + threadIdx.x;
    if (i < N_NODES) deg[i] = 1.0f / sqrtf(deg[i]);  // deg >= 1 always
}

// y[i,:] = dinv[i]^2 * x[i,:]   (self-loop term, initializes accumulator)
__global__ void sgc_seed(const float* __restrict__ xin, const float* __restrict__ dinv,
                         float* __restrict__ yout) {
    int t = blockIdx.x * blockDim.x + threadIdx.x;
    if (t < N_NODES * D_FEAT) {
        int i = t >> 7;
        float d = dinv[i];
        yout[t] = d * d * xin[t];
    }
}

// edge-parallel scatter: 32 lanes per edge, float4 per lane (128 feats)
__global__ void sgc_spmm(const float* __restrict__ xin, const int* __restrict__ src,
                         const int* __restrict__ dst, const float* __restrict__ dinv,
                         float* __restrict__ yout) {
    int t = blockIdx.x * blockDim.x + threadIdx.x;
    int e = t >> 5;
    int lane = t & 31;
    if (e >= N_EDGES) return;
    int s = src[e];
    int d = dst[e];
    float w = dinv[s] * dinv[d];
    const float4* xs = (const float4*)(xin + (size_t)s * D_FEAT) + lane;
    float4 v = *xs;
    float* yo = yout + (size_t)d * D_FEAT + lane * 4;
    atomicAdd(yo + 0, w * v.x);
    atomicAdd(yo + 1, w * v.y);
    atomicAdd(yo + 2, w * v.z);
    atomicAdd(yo + 3, w * v.w);
}

// ---------------- classifier: WMMA f32 16x16x4 GEMM + bias + log_softmax ----------------
// One wave handles 16 nodes x 48 classes (3 column tiles of 16, classes 40..47 masked).
// A layout (16x4 f32): lane m=lane&15, VGPR0/1 hold K = 2*(lane>>4) + {0,1}.
// B layout (4x16 f32): lane n=lane&15, VGPR0/1 hold K = 2*(lane>>4) + {0,1}  (B[k][n] = W[n][k]).
// C/D layout: VGPR r holds row (r + 8*(lane>>4)), column lane&15.

__global__ void sgc_gemm_lsm(const float* __restrict__ z, const float* __restrict__ W,
                             const float* __restrict__ bias, float* __restrict__ out) {
    int wave  = (blockIdx.x * blockDim.x + threadIdx.x) >> 5;
    int lane  = threadIdx.x & 31;
    int mbase = wave * 16;
    if (mbase >= N_NODES) return;          // wave-uniform: EXEC stays all-ones

    int half = lane >> 4;                  // which K pair / which row half
    int col  = lane & 15;
    int n0 = col, n1 = col + 16, n2 = col + 32;
    bool n2ok = (n2 < N_CLASSES);

    const float* arow  = z + (size_t)(mbase + col) * D_FEAT + half * 2;
    const float* brow0 = W + (size_t)n0 * D_FEAT + half * 2;
    const float* brow1 = W + (size_t)n1 * D_FEAT + half * 2;
    const float* brow2 = W + (size_t)(n2ok ? n2 : 0) * D_FEAT + half * 2;
    float mask2 = n2ok ? 1.0f : 0.0f;

    v8f acc0 = {}, acc1 = {}, acc2 = {};
    #pragma unroll
    for (int k = 0; k < D_FEAT; k += 4) {
        v2f a  = *(const v2f*)(arow + k);
        v2f b0 = *(const v2f*)(brow0 + k);
        v2f b1 = *(const v2f*)(brow1 + k);
        v2f b2 = *(const v2f*)(brow2 + k);
        b2 = b2 * mask2;  // zero padded classes
        acc0 = __builtin_amdgcn_wmma_f32_16x16x4_f32(false, a, false, b0, (short)0, acc0, false, false);
        acc1 = __builtin_amdgcn_wmma_f32_16x16x4_f32(false, a, false, b1, (short)0, acc1, false, false);
        acc2 = __builtin_amdgcn_wmma_f32_16x16x4_f32(false, a, false, b2, (short)0, acc2, false, false);
    }

    float bias0 = bias[n0];
    float bias1 = bias[n1];
    float bias2 = n2ok ? bias[n2] : 0.0f;

    #pragma unroll
    for (int r = 0; r < 8; ++r) {
        int orow = mbase + r + half * 8;
        float v0 = acc0[r] + bias0;
        float v1 = acc1[r] + bias1;
        float v2 = n2ok ? (acc2[r] + bias2) : -INFINITY;

        // row-wise max over the 16 lanes of this half (columns 0..47, masked -inf)
        float m = fmaxf(v0, fmaxf(v1, v2));
        #pragma unroll
        for (int off = 8; off >= 1; off >>= 1)
            m = fmaxf(m, __shfl_xor(m, off, 16));

        float s = expf(v0 - m) + expf(v1 - m) + (n2ok ? expf(v2 - m) : 0.0f);
        #pragma unroll
        for (int off = 8; off >= 1; off >>= 1)
            s += __shfl_xor(s, off, 16);

        float lse = m + logf(s);
        float* o = out + (size_t)orow * N_CLASSES;
        o[n0] = v0 - lse;
        o[n1] = v1 - lse;
        if (n2ok) o[n2] = v2 - lse;
    }
}

// ---------------- launch ----------------

extern "C" void kernel_launch(void* const* d_in, const int* in_sizes, int n_in,
                              void* d_out, int out_size, void* d_ws, size_t ws_size,
                              hipStream_t stream) {
    const float* x  = (const float*)d_in[0];
    const int*   ei = (const int*)d_in[1];      // [2, 600000] row-major
    const float* W  = (const float*)d_in[2];    // [40, 128]
    const float* b  = (const float*)d_in[3];    // [40]
    float* out = (float*)d_out;

    const int* src = ei;
    const int* dst = ei + N_EDGES;

    float* ws   = (float*)d_ws;
    float* dinv = ws;                                   // 100000 floats
    float* y    = ws + N_NODES;                         // 12.8M floats
    float* z    = y + (size_t)N_NODES * D_FEAT;         // 12.8M floats

    const int B = 256;

    sgc_deg_init <<<(N_NODES + B - 1) / B, B, 0, stream>>>(dinv);
    sgc_deg_accum<<<(N_EDGES + B - 1) / B, B, 0, stream>>>(dst, dinv);
    sgc_dinv     <<<(N_NODES + B - 1) / B, B, 0, stream>>>(dinv);

    // pass 1: y = Â x
    sgc_seed<<<(N_NODES * D_FEAT + B - 1) / B, B, 0, stream>>>(x, dinv, y);
    sgc_spmm<<<((size_t)N_EDGES * 32 + B - 1) / B, B, 0, stream>>>(x, src, dst, dinv, y);

    // pass 2: z = Â y
    sgc_seed<<<(N_NODES * D_FEAT + B - 1) / B, B, 0, stream>>>(y, dinv, z);
    sgc_spmm<<<((size_t)N_EDGES * 32 + B - 1) / B, B, 0, stream>>>(y, src, dst, dinv, z);

    // classifier + log_softmax: one wave per 16 nodes (100000 = 16 * 6250)
    int waves  = (N_NODES + 15) / 16;
    int blocks = (waves * 32 + B - 1) / B;
    sgc_gemm_lsm<<<blocks, B, 0, stream>>>(z, W, b, out);
}